// DirectTargetLoss_58050777972770
// MI455X (gfx1250) — compile-verified
//
#include <hip/hip_runtime.h>
#include <math.h>

// ---------------------------------------------------------------------------
// DirectTargetLoss for MI455X (gfx1250, wave32)
//   out[0] = -mean(log(tgt+eps))          over B*T gathered targets
//   out[1] =  mean(relu(1 - tgt))         over B*T
//   out[2] =  mean(top100 of each row with target slots excluded)
//
// Strategy: per-row radix-select (15-bit float-prefix histogram in LDS),
// two streaming passes (pass 2 hits L2: 131MB batch < 192MB L2), WMMA f32
// 16x16x4 for the final 256-element reductions.
// ---------------------------------------------------------------------------

#define B_SZ   256
#define V_SZ   128000
#define T_SZ   16
#define TOPK   100
#define EPSV   1e-8f
#define NBINS  (1 << 15)       // key = float bits [30:16]
#define CANDM  4096            // candidate buffer for threshold bin
#define THREADS 1024           // 32 waves per workgroup

typedef __attribute__((ext_vector_type(2))) float v2f;
typedef __attribute__((ext_vector_type(8))) float v8f;

#if defined(__has_builtin)
#  if __has_builtin(__builtin_amdgcn_wmma_f32_16x16x4_f32)
#    define HAVE_WMMA_F32X4 1
#  endif
#endif

__device__ __forceinline__ unsigned keyOf(float f) {
  return (__float_as_uint(f) >> 16) & 0x7FFFu;   // monotonic for f >= 0
}

// ---------------------------------------------------------------------------
// Kernel 1: gather target activations, per-row partial losses.
// ---------------------------------------------------------------------------
__global__ void gather_loss_kernel(const float* __restrict__ rep,
                                   const int* __restrict__ ids,
                                   float* __restrict__ tlRow,
                                   float* __restrict__ mlRow) {
  int row = blockIdx.x * blockDim.x + threadIdx.x;
  if (row >= B_SZ) return;
  const float* r = rep + (size_t)row * V_SZ;
  float tl = 0.f, ml = 0.f;
  for (int t = 0; t < T_SZ; ++t) {
    int id = ids[row * T_SZ + t];
    float v = r[id];
    tl += -logf(v + EPSV);
    float m = 1.0f - v;
    ml += (m > 0.f) ? m : 0.f;
  }
  tlRow[row] = tl;
  mlRow[row] = ml;
}

// ---------------------------------------------------------------------------
// Kernel 2: exact per-row top-100 sum via histogram radix-select.
// One workgroup (1024 thr = 32 waves) per row. Dynamic LDS ~152KB.
// ---------------------------------------------------------------------------
__global__ void __launch_bounds__(THREADS)
topk_row_kernel(const float* __restrict__ rep,
                const int* __restrict__ ids,
                float* __restrict__ nlRow) {
  const int row = blockIdx.x;
  const int tid = threadIdx.x;

  extern __shared__ unsigned char smemRaw[];
  unsigned* hist = (unsigned*)smemRaw;            // NBINS
  unsigned* csum = hist + NBINS;                  // THREADS chunk sums
  float*    cand = (float*)(csum + THREADS);      // CANDM
  float*    wVal = cand + CANDM;                  // 32 per-wave scratch
  int*      wIdx = (int*)(wVal + 32);             // 32
  int*      sTgt = wIdx + 32;                     // T_SZ target ids
  int*      sInt = sTgt + T_SZ;                   // [0]=bin [1]=need [2]=candCnt
  float*    sFlt = (float*)(sInt + 4);            // [0]=running sum

  // ---- init ----
  for (int i = tid; i < NBINS; i += THREADS) hist[i] = 0u;
  if (tid < T_SZ) sTgt[tid] = ids[row * T_SZ + tid];
  if (tid == 0) { sInt[2] = 0; sFlt[0] = 0.f; }
  __syncthreads();

  const float4* rp = (const float4*)(rep + (size_t)row * V_SZ);
  const int nvec = V_SZ / 4;

  // ---- pass A: histogram (stream from HBM, prefetch ahead) ----
  for (int i = tid; i < nvec; i += THREADS) {
    __builtin_prefetch(rp + i + 2 * THREADS, 0, 0);   // global_prefetch_b8
    float4 v = rp[i];
    atomicAdd(&hist[keyOf(v.x)], 1u);
    atomicAdd(&hist[keyOf(v.y)], 1u);
    atomicAdd(&hist[keyOf(v.z)], 1u);
    atomicAdd(&hist[keyOf(v.w)], 1u);
  }
  __syncthreads();

  // ---- remove (deduped) target positions from histogram ----
  if (tid == 0) {
    int uniq[T_SZ]; int nu = 0;
    for (int t = 0; t < T_SZ; ++t) {
      int id = sTgt[t];
      bool seen = false;
      for (int u = 0; u < nu; ++u) if (uniq[u] == id) { seen = true; break; }
      if (!seen) {
        uniq[nu++] = id;
        float v = rep[(size_t)row * V_SZ + id];
        hist[keyOf(v)] -= 1u;
      }
    }
  }
  __syncthreads();

  // ---- find threshold bin: top-down cumulative crossing of TOPK ----
  {
    constexpr int CHUNK = NBINS / THREADS;   // 32 bins per thread
    unsigned s = 0;
    const int base = tid * CHUNK;
    for (int j = 0; j < CHUNK; ++j) s += hist[base + j];
    csum[tid] = s;
  }
  __syncthreads();
  if (tid == 0) {
    constexpr int CHUNK = NBINS / THREADS;
    unsigned cum = 0; int b = 0; unsigned above = 0;
    for (int c = THREADS - 1; c >= 0; --c) {
      if (cum + csum[c] >= (unsigned)TOPK) {
        for (int bin = c * CHUNK + CHUNK - 1; bin >= c * CHUNK; --bin) {
          if (cum + hist[bin] >= (unsigned)TOPK) { b = bin; above = cum; break; }
          cum += hist[bin];
        }
        break;
      }
      cum += csum[c];
    }
    sInt[0] = b;
    sInt[1] = TOPK - (int)above;   // how many to take from bin b (1..100)
  }
  __syncthreads();

  const unsigned b = (unsigned)sInt[0];

  // ---- pass B: sum strictly-above bins + collect threshold-bin candidates
  //      (re-read hits L2: whole 131MB batch resident in 192MB L2) ----
  float localAbove = 0.f;
  for (int i = tid; i < nvec; i += THREADS) {
    float4 v = rp[i];
    const int gbase = i * 4;
    float vv[4] = { v.x, v.y, v.z, v.w };
#pragma unroll
    for (int c = 0; c < 4; ++c) {
      unsigned k = keyOf(vv[c]);
      if (k >= b) {                         // rare path (~a few hundred / row)
        int gidx = gbase + c;
        bool isT = false;
#pragma unroll
        for (int t = 0; t < T_SZ; ++t) isT |= (sTgt[t] == gidx);
        if (!isT) {
          if (k > b) {
            localAbove += vv[c];
          } else {
            int pos = atomicAdd(&sInt[2], 1);
            if (pos < CANDM) cand[pos] = vv[c];
          }
        }
      }
    }
  }

  // block reduction of localAbove (wave32 shuffles -> LDS -> thread 0)
  for (int off = 16; off > 0; off >>= 1)
    localAbove += __shfl_down(localAbove, off, 32);
  if ((tid & 31) == 0) wVal[tid >> 5] = localAbove;
  __syncthreads();
  if (tid == 0) {
    float s = 0.f;
    for (int w = 0; w < THREADS / 32; ++w) s += wVal[w];
    sFlt[0] = s;
  }
  __syncthreads();

  // ---- select 'need' largest from the threshold bin (repeated arg-max) ----
  const int cnt  = (sInt[2] < CANDM) ? sInt[2] : CANDM;
  const int need = sInt[1];
  for (int it = 0; it < need; ++it) {
    float best = -3.4e38f; int bi = -1;
    for (int j = tid; j < cnt; j += THREADS) {
      float v = cand[j];
      if (v > best) { best = v; bi = j; }
    }
    for (int off = 16; off > 0; off >>= 1) {
      float ov = __shfl_down(best, off, 32);
      int   oi = __shfl_down(bi,   off, 32);
      if (ov > best) { best = ov; bi = oi; }
    }
    if ((tid & 31) == 0) { wVal[tid >> 5] = best; wIdx[tid >> 5] = bi; }
    __syncthreads();
    if (tid == 0) {
      float bb = -3.4e38f; int bj = -1;
      for (int w = 0; w < THREADS / 32; ++w)
        if (wVal[w] > bb) { bb = wVal[w]; bj = wIdx[w]; }
      if (bj >= 0) { sFlt[0] += bb; cand[bj] = -3.4e38f; }
    }
    __syncthreads();
  }

  if (tid == 0) nlRow[row] = sFlt[0];
}

// ---------------------------------------------------------------------------
// Kernel 3: reduce the three 256-element partial arrays on the matrix pipe.
// Sum(256) = 4 chained v_wmma_f32_16x16x4_f32 with B = ones:
//   D[m][n] = sum_k A[m][k]  ->  per-lane sum of the 8 acc VGPRs gives the
//   M=0..7 total in lanes 0..15 and M=8..15 total in lanes 16..31.
// Single full wave32, uniform control flow (EXEC all ones, as WMMA requires).
// ---------------------------------------------------------------------------
__device__ __forceinline__ float wmma_sum256(const float* __restrict__ p, int lane) {
#if defined(HAVE_WMMA_F32X4)
  v8f acc = {};
  const int m  = lane & 15;
  const int k0 = (lane & 16) ? 2 : 0;   // A 16x4 layout: lanes 0-15 K=0,1; 16-31 K=2,3
  v2f ones; ones.x = 1.0f; ones.y = 1.0f;
#pragma unroll
  for (int c = 0; c < 4; ++c) {
    v2f a;
    a.x = p[c * 64 + m * 4 + k0];
    a.y = p[c * 64 + m * 4 + k0 + 1];
    acc = __builtin_amdgcn_wmma_f32_16x16x4_f32(
        /*neg_a=*/false, a, /*neg_b=*/false, ones,
        /*c_mod=*/(short)0, acc, /*reuse_a=*/false, /*reuse_b=*/false);
  }
  float s = 0.f;
#pragma unroll
  for (int r = 0; r < 8; ++r) s += acc[r];
  // lanes 0-15 hold sum over M=0..7, lanes 16-31 hold sum over M=8..15 (any N)
  return __shfl(s, 0, 32) + __shfl(s, 16, 32);
#else
  // VALU fallback: strided accumulate + wave32 shuffle reduce
  float s = 0.f;
  for (int i = lane; i < 256; i += 32) s += p[i];
  for (int off = 16; off > 0; off >>= 1) s += __shfl_down(s, off, 32);
  return __shfl(s, 0, 32);
#endif
}

__global__ void finalize_kernel(const float* __restrict__ tlRow,
                                const float* __restrict__ mlRow,
                                const float* __restrict__ nlRow,
                                float* __restrict__ out) {
  const int lane = threadIdx.x;
  float t = wmma_sum256(tlRow, lane);
  float m = wmma_sum256(mlRow, lane);
  float n = wmma_sum256(nlRow, lane);
  if (lane == 0) {
    out[0] = t / (float)(B_SZ * T_SZ);
    out[1] = m / (float)(B_SZ * T_SZ);
    out[2] = n / (float)(B_SZ * TOPK);
  }
}

// ---------------------------------------------------------------------------
// Host launch
// ---------------------------------------------------------------------------
static const size_t SMEM_BYTES =
    (size_t)(NBINS + THREADS + CANDM + 32 + 32 + T_SZ + 8) * sizeof(unsigned);

extern "C" void kernel_launch(void* const* d_in, const int* in_sizes, int n_in,
                              void* d_out, int out_size, void* d_ws, size_t ws_size,
                              hipStream_t stream) {
  (void)in_sizes; (void)n_in; (void)out_size; (void)ws_size;

  const float* rep = (const float*)d_in[0];
  const int*   ids = (const int*)d_in[1];
  float* out = (float*)d_out;

  float* ws    = (float*)d_ws;
  float* tlRow = ws;              // 256
  float* mlRow = ws + B_SZ;       // 256
  float* nlRow = ws + 2 * B_SZ;   // 256

  // allow >64KB dynamic LDS (CDNA5 WGP has 320KB)
  hipFuncSetAttribute(reinterpret_cast<const void*>(topk_row_kernel),
                      hipFuncAttributeMaxDynamicSharedMemorySize,
                      (int)SMEM_BYTES);

  gather_loss_kernel<<<1, B_SZ, 0, stream>>>(rep, ids, tlRow, mlRow);
  topk_row_kernel<<<B_SZ, THREADS, SMEM_BYTES, stream>>>(rep, ids, nlRow);
  finalize_kernel<<<1, 32, 0, stream>>>(tlRow, mlRow, nlRow, out);
}